// Cell_74552042324717
// MI455X (gfx1250) — compile-verified
//
#include <hip/hip_runtime.h>
#include <math.h>

// ---------------- constants ----------------
static constexpr int Bn   = 32;            // batch
static constexpr int Cc   = 96;            // channels per state
static constexpr int HW   = 1024;          // 32*32
static constexpr int PB   = Cc * HW;       // 98304 elems / batch (standard stride)
static constexpr int TOT  = Bn * PB;       // 3145728 elems / tensor
static constexpr long BSO = 384 * 1024;    // batch stride inside concatenated d_out

typedef float v2f __attribute__((ext_vector_type(2)));
typedef float v8f __attribute__((ext_vector_type(8)));

#define GUARD(gidx, e, op) do { if ((e) >= 0 && (gidx)[(e)] != (op)) return; } while (0)

// ---------------- utility ----------------
__global__ void zero_f32(float* p, int n) {
  int i = blockIdx.x * blockDim.x + threadIdx.x;
  if (i < n) p[i] = 0.0f;
}

// ---------------- 1x1 conv as fp32 WMMA GEMM ----------------
// D[m][n] = sum_k W[m][k] * X[k][n],  M=96, K=96, N=32768 (n = b*1024 + hw)
// One wave computes a full 96x16 strip (6 tiles of 16x16), K stepped by 4 via
// v_wmma_f32_16x16x4_f32. Activation column loaded once per K-step, reused 6x.
__global__ __launch_bounds__(128) void conv1x1_wmma(
    const float* __restrict__ X, long bsX, int reluIn,
    const float* __restrict__ Wm,
    float* __restrict__ Y, long bsY,
    const int* __restrict__ gidx, int e, int op)
{
  GUARD(gidx, e, op);
  const int lane  = threadIdx.x & 31;
  const int wv    = threadIdx.x >> 5;
  const int ntile = blockIdx.x * 4 + wv;          // 2048 n-tiles total
  const int lm    = lane & 15;
  const int n     = ntile * 16 + lm;
  const int kseg  = (lane >> 4) << 1;             // 0 or 2
  const int b     = n >> 10;
  const int hw    = n & 1023;
  const float* xcol = X + (long)b * bsX + hw;

  v8f acc[6] = {};
  for (int k = 0; k < Cc; k += 4) {
    float b0 = xcol[(k + kseg) << 10];
    float b1 = xcol[(k + kseg + 1) << 10];
    if (reluIn) { b0 = fmaxf(b0, 0.0f); b1 = fmaxf(b1, 0.0f); }
    v2f bv; bv.x = b0; bv.y = b1;
#pragma unroll
    for (int mt = 0; mt < 6; mt++) {
      const float* wr = Wm + (mt * 16 + lm) * Cc + k + kseg;
      v2f av; av.x = wr[0]; av.y = wr[1];
      acc[mt] = __builtin_amdgcn_wmma_f32_16x16x4_f32(
          false, av, false, bv, (short)0, acc[mt], false, false);
    }
  }
  const int mo = (lane >> 4) << 3;                // +8 for upper lane half
#pragma unroll
  for (int mt = 0; mt < 6; mt++) {
#pragma unroll
    for (int r = 0; r < 8; r++) {
      Y[(long)b * bsY + (mt * 16 + mo + r) * HW + hw] = acc[mt][r];
    }
  }
}

// ---------------- BatchNorm (affine=False, batch stats) ----------------
__global__ __launch_bounds__(256) void bn_stats(
    const float* __restrict__ X, long bsX, float* __restrict__ stats,
    const int* __restrict__ gidx, int e, int op)
{
  GUARD(gidx, e, op);
  const int c = blockIdx.x;
  float s = 0.0f, sq = 0.0f;
  for (int i = threadIdx.x; i < Bn * HW; i += 256) {
    int b = i >> 10, hw = i & 1023;
    float v = X[(long)b * bsX + c * HW + hw];
    s += v; sq += v * v;
  }
  __shared__ float sh[256], shq[256];
  sh[threadIdx.x] = s; shq[threadIdx.x] = sq;
  __syncthreads();
  for (int st = 128; st > 0; st >>= 1) {
    if (threadIdx.x < st) {
      sh[threadIdx.x]  += sh[threadIdx.x + st];
      shq[threadIdx.x] += shq[threadIdx.x + st];
    }
    __syncthreads();
  }
  if (threadIdx.x == 0) {
    const float inv = 1.0f / (float)(Bn * HW);
    float mean = sh[0] * inv;
    float var  = shq[0] * inv - mean * mean;
    stats[c]      = mean;
    stats[Cc + c] = 1.0f / sqrtf(var + 1e-5f);
  }
}

__global__ void bn_apply(
    const float* __restrict__ X, long bsX, const float* __restrict__ stats,
    float* __restrict__ Y, long bsY,
    const int* __restrict__ gidx, int e, int op)
{
  GUARD(gidx, e, op);
  int i = blockIdx.x * blockDim.x + threadIdx.x;
  if (i >= TOT) return;
  int b = i / PB, rem = i % PB, c = rem >> 10, hw = rem & 1023;
  float v = X[(long)b * bsX + c * HW + hw];
  Y[(long)b * bsY + c * HW + hw] = (v - stats[c]) * stats[Cc + c];
}

// ---------------- depthwise conv (optional input ReLU, dilation) ----------------
__global__ void dwconv(
    const float* __restrict__ X, long bsX, const float* __restrict__ Wt,
    int k, int dil, int pad, int reluIn,
    float* __restrict__ Y, long bsY,
    const int* __restrict__ gidx, int e, int op)
{
  GUARD(gidx, e, op);
  int i = blockIdx.x * blockDim.x + threadIdx.x;
  if (i >= TOT) return;
  int b = i / PB, rem = i % PB, c = rem >> 10, hw = rem & 1023;
  int y = hw >> 5, x = hw & 31;
  const float* xb = X + (long)b * bsX + c * HW;
  const float* wc = Wt + c * k * k;
  float acc = 0.0f;
  for (int ky = 0; ky < k; ky++) {
    int yy = y + ky * dil - pad;
    if (yy < 0 || yy >= 32) continue;
    for (int kx = 0; kx < k; kx++) {
      int xx = x + kx * dil - pad;
      if (xx < 0 || xx >= 32) continue;
      float v = xb[yy * 32 + xx];
      if (reluIn) v = fmaxf(v, 0.0f);
      acc += wc[ky * k + kx] * v;
    }
  }
  Y[(long)b * bsY + c * HW + hw] = acc;
}

// ---------------- 3x3 pools, stride 1, pad 1 ----------------
__global__ void maxpool3(
    const float* __restrict__ X, long bsX, float* __restrict__ Y, long bsY,
    const int* __restrict__ gidx, int e, int op)
{
  GUARD(gidx, e, op);
  int i = blockIdx.x * blockDim.x + threadIdx.x;
  if (i >= TOT) return;
  int b = i / PB, rem = i % PB, c = rem >> 10, hw = rem & 1023;
  int y = hw >> 5, x = hw & 31;
  const float* xb = X + (long)b * bsX + c * HW;
  float m = -INFINITY;
  for (int dy = -1; dy <= 1; dy++) {
    int yy = y + dy; if (yy < 0 || yy >= 32) continue;
    for (int dx = -1; dx <= 1; dx++) {
      int xx = x + dx; if (xx < 0 || xx >= 32) continue;
      m = fmaxf(m, xb[yy * 32 + xx]);
    }
  }
  Y[(long)b * bsY + c * HW + hw] = m;
}

__global__ void avgpool3(
    const float* __restrict__ X, long bsX, float* __restrict__ Y, long bsY,
    const int* __restrict__ gidx, int e, int op)
{
  GUARD(gidx, e, op);
  int i = blockIdx.x * blockDim.x + threadIdx.x;
  if (i >= TOT) return;
  int b = i / PB, rem = i % PB, c = rem >> 10, hw = rem & 1023;
  int y = hw >> 5, x = hw & 31;
  const float* xb = X + (long)b * bsX + c * HW;
  float s = 0.0f; int cnt = 0;
  for (int dy = -1; dy <= 1; dy++) {
    int yy = y + dy; if (yy < 0 || yy >= 32) continue;
    for (int dx = -1; dx <= 1; dx++) {
      int xx = x + dx; if (xx < 0 || xx >= 32) continue;
      s += xb[yy * 32 + xx]; cnt++;
    }
  }
  Y[(long)b * bsY + c * HW + hw] = s / (float)cnt;
}

// ---------------- edge combine: out += w[idx]*sel + (sum(w) - w[idx]) ----------------
__global__ void edge_accum(
    float* __restrict__ out, long bsOut,
    const float* __restrict__ br, long bsBr,
    const float* __restrict__ sj, long bsSj,
    const float* __restrict__ hardwts, const int* __restrict__ gidx, int e)
{
  int i = blockIdx.x * blockDim.x + threadIdx.x;
  if (i >= TOT) return;
  const int idx = gidx[e];
  const float w = hardwts[e * 8 + idx];
  float sum = 0.0f;
  for (int o = 0; o < 8; o++) sum += hardwts[e * 8 + o];
  const float bias = sum - w;
  int b = i / PB, rem = i % PB, c = rem >> 10, hw = rem & 1023;
  float val;
  if (idx == 0)      val = 0.0f;                                  // none
  else if (idx == 3) val = sj[(long)b * bsSj + c * HW + hw];      // skip
  else               val = br[(long)b * bsBr + c * HW + hw];      // computed branch
  out[(long)b * bsOut + c * HW + hw] += w * val + bias;
}

// ---------------- host orchestration ----------------
extern "C" void kernel_launch(void* const* d_in, const int* in_sizes, int n_in,
                              void* d_out, int out_size, void* d_ws, size_t ws_size,
                              hipStream_t stream)
{
  (void)ws_size;
  // ---- locate inputs robustly by element counts ----
  int i_s0 = -1, i_s1 = -1, i_hw = -1, i_idx = -1, firstEdge = -1;
  for (int i = 0; i < n_in; i++) {
    const int sz = in_sizes[i];
    if (sz == TOT) { if (i_s0 < 0) i_s0 = i; else if (i_s1 < 0) i_s1 = i; }
    else if (sz == 112) i_hw = i;
    else if (sz == 14)  i_idx = i;
    if ((sz == 864 || sz == 2400) && firstEdge < 0) firstEdge = i;
  }
  if (i_s0 < 0 || i_s1 < 0 || i_hw < 0 || i_idx < 0 || firstEdge < 0) return;
  int i_pre0 = -1, i_pre1 = -1;
  for (int i = 0; i < n_in; i++) {
    if (in_sizes[i] == 9216 && (i < firstEdge || i >= firstEdge + 168)) {
      if (i_pre0 < 0) i_pre0 = i; else if (i_pre1 < 0) i_pre1 = i;
    }
  }
  if (i_pre0 < 0 || i_pre1 < 0) return;

  // per-edge sub-array offsets: detect insertion-order vs sorted-tree flatten
  int o_s3dw1, o_s3pw1, o_s3dw2, o_s3pw2, o_s5dw1, o_s5pw1, o_s5dw2, o_s5pw2;
  int o_d3dw, o_d3pw, o_d5dw, o_d5pw;
  const bool sepFirst = (firstEdge + 2 < n_in) && (in_sizes[firstEdge + 2] == 864);
  if (sepFirst) { // sep3{dw1,pw1,dw2,pw2}, sep5{...}, dil3{dw,pw}, dil5{dw,pw}
    o_s3dw1=0;o_s3pw1=1;o_s3dw2=2;o_s3pw2=3;o_s5dw1=4;o_s5pw1=5;o_s5dw2=6;o_s5pw2=7;
    o_d3dw=8;o_d3pw=9;o_d5dw=10;o_d5pw=11;
  } else {        // dil3{dw,pw}, dil5{dw,pw}, sep3{dw1,dw2,pw1,pw2}, sep5{...}
    o_d3dw=0;o_d3pw=1;o_d5dw=2;o_d5pw=3;
    o_s3dw1=4;o_s3dw2=5;o_s3pw1=6;o_s3pw2=7;o_s5dw1=8;o_s5dw2=9;o_s5pw1=10;o_s5pw2=11;
  }

  const float* s0      = (const float*)d_in[i_s0];
  const float* s1      = (const float*)d_in[i_s1];
  const float* hardwts = (const float*)d_in[i_hw];
  const int*   gidx    = (const int*)d_in[i_idx];
  const float* pre0w   = (const float*)d_in[i_pre0];
  const float* pre1w   = (const float*)d_in[i_pre1];
  auto EW = [&](int e, int off) { return (const float*)d_in[firstEdge + e * 12 + off]; };

  float* ws    = (float*)d_ws;
  float* pre0  = ws;
  float* pre1  = pre0 + TOT;
  float* t0    = pre1 + TOT;
  float* t1    = t0 + TOT;
  float* br    = t1 + TOT;
  float* stats = br + TOT;   // 192 floats
  float* out   = (float*)d_out;

  const long BS = PB;
  const dim3 gEW((TOT + 255) / 256), bEW(256);
  const dim3 gGE(512), bGE(128);     // 2048 n-tiles, 4 waves/block

  zero_f32<<<(out_size + 255) / 256, 256, 0, stream>>>(out, out_size);

  // preprocess: s = BN(conv1x1(relu(s), w))
  conv1x1_wmma<<<gGE, bGE, 0, stream>>>(s0, BS, 1, pre0w, t0, BS, gidx, -1, 0);
  bn_stats<<<Cc, 256, 0, stream>>>(t0, BS, stats, gidx, -1, 0);
  bn_apply<<<gEW, bEW, 0, stream>>>(t0, BS, stats, pre0, BS, gidx, -1, 0);
  conv1x1_wmma<<<gGE, bGE, 0, stream>>>(s1, BS, 1, pre1w, t0, BS, gidx, -1, 0);
  bn_stats<<<Cc, 256, 0, stream>>>(t0, BS, stats, gidx, -1, 0);
  bn_apply<<<gEW, bEW, 0, stream>>>(t0, BS, stats, pre1, BS, gidx, -1, 0);

  struct View { float* p; long bs; } st[6];
  st[0] = { pre0, BS }; st[1] = { pre1, BS };
  for (int nd = 2; nd <= 5; nd++) st[nd] = { out + (long)(nd - 2) * Cc * HW, BSO };

  int e = 0;
  for (int node = 2; node <= 5; node++) {
    for (int j = 0; j < node; j++, e++) {
      const float* x = st[j].p; const long bsx = st[j].bs;
      // op 1/2: pools (guarded)
      maxpool3<<<gEW, bEW, 0, stream>>>(x, bsx, br, BS, gidx, e, 1);
      avgpool3<<<gEW, bEW, 0, stream>>>(x, bsx, br, BS, gidx, e, 2);
      // ops 4/5: separable convs (guarded pipelines)
      {
        const int ks[2] = {3, 5}, ops[2] = {4, 5};
        const int offs[2][4] = {{o_s3dw1, o_s3pw1, o_s3dw2, o_s3pw2},
                                {o_s5dw1, o_s5pw1, o_s5dw2, o_s5pw2}};
        for (int t = 0; t < 2; t++) {
          const int k = ks[t], op = ops[t], pad = k / 2;
          dwconv<<<gEW, bEW, 0, stream>>>(x, bsx, EW(e, offs[t][0]), k, 1, pad, 1, t0, BS, gidx, e, op);
          conv1x1_wmma<<<gGE, bGE, 0, stream>>>(t0, BS, 0, EW(e, offs[t][1]), t1, BS, gidx, e, op);
          bn_stats<<<Cc, 256, 0, stream>>>(t1, BS, stats, gidx, e, op);
          bn_apply<<<gEW, bEW, 0, stream>>>(t1, BS, stats, t0, BS, gidx, e, op);
          dwconv<<<gEW, bEW, 0, stream>>>(t0, BS, EW(e, offs[t][2]), k, 1, pad, 1, t1, BS, gidx, e, op);
          conv1x1_wmma<<<gGE, bGE, 0, stream>>>(t1, BS, 0, EW(e, offs[t][3]), t0, BS, gidx, e, op);
          bn_stats<<<Cc, 256, 0, stream>>>(t0, BS, stats, gidx, e, op);
          bn_apply<<<gEW, bEW, 0, stream>>>(t0, BS, stats, br, BS, gidx, e, op);
        }
      }
      // ops 6/7: dilated convs (guarded pipelines)
      {
        const int ks[2] = {3, 5}, ops[2] = {6, 7};
        const int offs[2][2] = {{o_d3dw, o_d3pw}, {o_d5dw, o_d5pw}};
        for (int t = 0; t < 2; t++) {
          const int k = ks[t], op = ops[t], pad = (k / 2) * 2;
          dwconv<<<gEW, bEW, 0, stream>>>(x, bsx, EW(e, offs[t][0]), k, 2, pad, 1, t0, BS, gidx, e, op);
          conv1x1_wmma<<<gGE, bGE, 0, stream>>>(t0, BS, 0, EW(e, offs[t][1]), t1, BS, gidx, e, op);
          bn_stats<<<Cc, 256, 0, stream>>>(t1, BS, stats, gidx, e, op);
          bn_apply<<<gEW, bEW, 0, stream>>>(t1, BS, stats, br, BS, gidx, e, op);
        }
      }
      edge_accum<<<gEW, bEW, 0, stream>>>(st[node].p, st[node].bs, br, BS, x, bsx, hardwts, gidx, e);
    }
  }
}